// ConvEdgeType_20383914787325
// MI455X (gfx1250) — compile-verified
//
#include <hip/hip_runtime.h>

#define IN_CH   128
#define OUT_CH  128
#define N_TYPES 4
#define NNODES  50000
#define NEDGES  500000

typedef __attribute__((ext_vector_type(2))) float v2f;
typedef __attribute__((ext_vector_type(8))) float v8f;

// ---------------------------------------------------------------------------
// Precompute effective weights:
//   slice s in [0,8): WeffT[s][n][k] = sum_j Wl_s[k][j] * Wlin[s*128+j][n]
//   slice 8:          WeffXT[n][k]   = sum_s sum_j Wr_s[k][j] * Wlin[s*128+j][n]
// (stored transposed, n-major, so WMMA B-operand loads match A-operand layout)
// ---------------------------------------------------------------------------
__global__ void precompute_weights(const float* __restrict__ WAl, const float* __restrict__ WBl,
                                   const float* __restrict__ WAr, const float* __restrict__ WBr,
                                   const float* __restrict__ Wlin,
                                   float* __restrict__ WeffT, float* __restrict__ WeffXT) {
    int k = blockIdx.x;      // 0..127 (input channel)
    int slice = blockIdx.y;  // 0..8
    int n = threadIdx.x;     // 0..127 (output channel)
    if (slice < 8) {
        int dir = slice & 1, t = slice >> 1;
        const float* Wl = (dir ? WBl : WAl) + (long)t * IN_CH * OUT_CH + (long)k * OUT_CH;
        float acc = 0.f;
        for (int j = 0; j < OUT_CH; ++j)
            acc += Wl[j] * Wlin[(long)(slice * OUT_CH + j) * OUT_CH + n];
        WeffT[(long)slice * IN_CH * OUT_CH + (long)n * IN_CH + k] = acc;
    } else {
        float acc = 0.f;
        for (int s = 0; s < 8; ++s) {
            int dir = s & 1, t = s >> 1;
            const float* Wr = (dir ? WBr : WAr) + (long)t * IN_CH * OUT_CH + (long)k * OUT_CH;
            for (int j = 0; j < OUT_CH; ++j)
                acc += Wr[j] * Wlin[(long)(s * OUT_CH + j) * OUT_CH + n];
        }
        WeffXT[(long)n * IN_CH + k] = acc;
    }
}

__global__ void precompute_bias(const float* __restrict__ bA, const float* __restrict__ bB,
                                const float* __restrict__ blin, const float* __restrict__ Wlin,
                                float* __restrict__ beff) {
    int n = threadIdx.x;
    float acc = blin[n];
    for (int s = 0; s < 8; ++s) {
        const float* b = ((s & 1) ? bB : bA) + (s >> 1) * OUT_CH;
        for (int j = 0; j < OUT_CH; ++j)
            acc += b[j] * Wlin[(long)(s * OUT_CH + j) * OUT_CH + n];
    }
    beff[n] = acc;
}

// ---------------------------------------------------------------------------
// Zero the (L2-resident) aggregation buffers.
// ---------------------------------------------------------------------------
__global__ void zero_kernel(float4* __restrict__ p, long n4) {
    long i = (long)blockIdx.x * blockDim.x + threadIdx.x;
    long stride = (long)gridDim.x * blockDim.x;
    float4 z = make_float4(0.f, 0.f, 0.f, 0.f);
    for (; i < n4; i += stride) p[i] = z;
}

// ---------------------------------------------------------------------------
// Per-edge-type scatter: one wave32 per edge; lanes cover 128 channels in 4
// steps. x (25.6 MB) and agg buffers (51.2 MB) stay in the 192 MB L2, so the
// random gathers + atomic adds never hit HBM.
// ---------------------------------------------------------------------------
__global__ void scatter_kernel(const float* __restrict__ x,
                               const long long* __restrict__ ei,
                               const long long* __restrict__ et,
                               float* __restrict__ aggA, float* __restrict__ aggB,
                               int t, int nE) {
    int gid = blockIdx.x * blockDim.x + threadIdx.x;
    int wave = gid >> 5;
    int lane = gid & 31;
    int nw = (gridDim.x * blockDim.x) >> 5;
    for (int e = wave; e < nE; e += nw) {
        long long s = ei[e];
        long long d = ei[nE + e];
        if (et[e] != (long long)t || s == d) continue;
        const float* xs = x + s * IN_CH;
        const float* xd = x + d * IN_CH;
        float* pa = aggA + d * IN_CH;
        float* pb = aggB + s * IN_CH;
#pragma unroll
        for (int i = 0; i < 4; ++i) {
            int c = lane + i * 32;
            atomicAdd(pa + c, xs[c]);
            atomicAdd(pb + c, xd[c]);
        }
    }
}

// ---------------------------------------------------------------------------
// Fused WMMA GEMM: out_tile (+)= A1@B1^T + A2@B2^T [+ A3@B3^T + bias]
// A matrices are [N,128] row-major; B matrices are effective weights stored
// transposed [128(n)][128(k)]. Block = 128 threads = 4 wave32; each wave owns
// one 16x16 N-tile, block tile = 16 nodes x 64 cols, gridDim.y=2 covers 128.
// f32 WMMA operand layout: lane L reads float2 at row (L%16), k = k0+2*(L/16),
// identical for A and transposed-B. LDS pitch 130 floats -> 8B-aligned,
// bank-conflict-free ds_load_b64.
// ---------------------------------------------------------------------------
__launch_bounds__(128)
__global__ void gemm_kernel(const float* __restrict__ A1, const float* __restrict__ B1,
                            const float* __restrict__ A2, const float* __restrict__ B2,
                            const float* __restrict__ A3, const float* __restrict__ B3,
                            const float* __restrict__ bias,
                            float* __restrict__ out, int firstPass) {
    __shared__ float ldsA[16 * 130];
    __shared__ float ldsB[64 * 130];
    const int tid = threadIdx.x;
    const int lane = tid & 31;
    const int wv = tid >> 5;
    const int node0 = blockIdx.x * 16;
    const int colbase = blockIdx.y * 64;
    const int m = lane & 15;
    const int kh = (lane >> 4) * 2;

    v8f acc = {0.f, 0.f, 0.f, 0.f, 0.f, 0.f, 0.f, 0.f};

    const float* As[3] = {A1, A2, A3};
    const float* Bs[3] = {B1, B2, B3};

    for (int g = 0; g < 3; ++g) {
        const float* A = As[g];
        if (!A) break;
        const float* Bt = Bs[g];
        // cooperative, coalesced tile loads
        for (int idx = tid; idx < 16 * IN_CH; idx += 128) {
            int r = idx >> 7, c = idx & 127;
            ldsA[r * 130 + c] = A[(long)(node0 + r) * IN_CH + c];
        }
        for (int idx = tid; idx < 64 * IN_CH; idx += 128) {
            int r = idx >> 7, c = idx & 127;
            ldsB[r * 130 + c] = Bt[(long)(colbase + r) * IN_CH + c];
        }
        __syncthreads();

        const float* pa = &ldsA[m * 130 + kh];
        const float* pb = &ldsB[(wv * 16 + m) * 130 + kh];
#pragma unroll
        for (int k0 = 0; k0 < IN_CH; k0 += 4) {
            v2f a = *(const v2f*)(pa + k0);
            v2f b = *(const v2f*)(pb + k0);
            acc = __builtin_amdgcn_wmma_f32_16x16x4_f32(
                false, a, false, b, (short)0, acc, false, false);
        }
        __syncthreads();
    }

    const int col = colbase + wv * 16 + m;
    if (bias) {
        float bv = bias[col];
#pragma unroll
        for (int r = 0; r < 8; ++r) acc[r] += bv;
    }
    // C/D layout: VGPR r holds M=r (lanes 0-15) and M=r+8 (lanes 16-31)
    const int rowbase = node0 + ((lane >> 4) << 3);
#pragma unroll
    for (int r = 0; r < 8; ++r) {
        long idx = (long)(rowbase + r) * OUT_CH + col;
        if (firstPass) out[idx] = acc[r];
        else           out[idx] += acc[r];
    }
}

// ---------------------------------------------------------------------------
extern "C" void kernel_launch(void* const* d_in, const int* in_sizes, int n_in,
                              void* d_out, int out_size, void* d_ws, size_t ws_size,
                              hipStream_t stream) {
    (void)in_sizes; (void)n_in; (void)out_size; (void)ws_size;

    const float*     x    = (const float*)d_in[0];
    const long long* ei   = (const long long*)d_in[1];
    const long long* et   = (const long long*)d_in[2];
    const float*     WAl  = (const float*)d_in[3];
    const float*     WAr  = (const float*)d_in[4];
    const float*     bA   = (const float*)d_in[5];
    const float*     WBl  = (const float*)d_in[6];
    const float*     WBr  = (const float*)d_in[7];
    const float*     bB   = (const float*)d_in[8];
    const float*     Wlin = (const float*)d_in[9];
    const float*     blin = (const float*)d_in[10];
    float* out = (float*)d_out;

    float* ws     = (float*)d_ws;
    float* WeffT  = ws;                                      // 8*128*128
    float* WeffXT = WeffT + 8L * IN_CH * OUT_CH;             // 128*128
    float* beff   = WeffXT + (long)IN_CH * OUT_CH;           // 128 (+pad)
    float* aggA   = beff + 256;                              // 50000*128
    float* aggB   = aggA + (long)NNODES * IN_CH;             // 50000*128

    // Effective weights / bias (folds concat + final linear into per-type GEMMs)
    precompute_weights<<<dim3(128, 9), 128, 0, stream>>>(WAl, WBl, WAr, WBr, Wlin,
                                                         WeffT, WeffXT);
    precompute_bias<<<1, 128, 0, stream>>>(bA, bB, blin, Wlin, beff);

    const long agg_floats = 2L * NNODES * IN_CH;
    for (int t = 0; t < N_TYPES; ++t) {
        zero_kernel<<<2048, 256, 0, stream>>>((float4*)aggA, agg_floats / 4);
        scatter_kernel<<<4096, 256, 0, stream>>>(x, ei, et, aggA, aggB, t, NEDGES);

        const float* B1 = WeffT + (long)(2 * t + 0) * IN_CH * OUT_CH;
        const float* B2 = WeffT + (long)(2 * t + 1) * IN_CH * OUT_CH;
        if (t == 0) {
            // out = aggA@B1 + aggB@B2 + x@WeffX + b_eff
            gemm_kernel<<<dim3(NNODES / 16, 2), 128, 0, stream>>>(
                aggA, B1, aggB, B2, x, WeffXT, beff, out, 1);
        } else {
            // out += aggA@B1 + aggB@B2
            gemm_kernel<<<dim3(NNODES / 16, 2), 128, 0, stream>>>(
                aggA, B1, aggB, B2, nullptr, nullptr, nullptr, out, 0);
        }
    }
}